// ModelNew_1580547966550
// MI455X (gfx1250) — compile-verified
//
#include <hip/hip_runtime.h>
#include <cstdint>
#include <cstddef>

#define LSTM_H 512
#define LSTM_B 16
#define LSTM_T 1024
#define G4H    2048
#define NWG    32          // recurrent workgroups = 512/16 h-column tiles

typedef float v2f __attribute__((ext_vector_type(2)));
typedef float v8f __attribute__((ext_vector_type(8)));

__device__ __forceinline__ float sigmoidf_(float x) {
  return 1.0f / (1.0f + __expf(-x));
}

// ---------------------------------------------------------------------------
// Projection GEMM: gx[M][N] = X[M][K=512] * W[N][K]^T + (b1[N] + b2[N])
// M = B*T = 16384, N = 4H = 2048.  64x64 block tile, K chunked by 256,
// 8 waves/WG, each wave owns two 16x16 WMMA tiles (v_wmma_f32_16x16x4_f32)
// giving two independent accumulator chains.
// ---------------------------------------------------------------------------
__global__ __launch_bounds__(256) void proj_gemm(
    const float* __restrict__ X, const float* __restrict__ W,
    const float* __restrict__ b1, const float* __restrict__ b2,
    float* __restrict__ gx)
{
  __shared__ float sA[64 * 260];   // pad 4 dwords -> conflict-free b64 reads
  __shared__ float sB[64 * 260];

  const int tid  = threadIdx.x;
  const int lane = tid & 31;
  const int wave = tid >> 5;
  const int l16  = lane & 15;
  const int hi   = lane >> 4;              // half-wave: K sub-pair select
  const int mBase = blockIdx.y * 64;
  const int nBase = blockIdx.x * 64;

  const int mi0 = wave >> 2,       ni0 = wave & 3;        // tile ids 0..7
  const int mi1 = (wave + 8) >> 2, ni1 = (wave + 8) & 3;  // tile ids 8..15

  v8f acc0 = {}; v8f acc1 = {};

  for (int kc = 0; kc < 512; kc += 256) {
    __syncthreads();
    for (int i = tid; i < 64 * 64; i += 256) {   // 64 rows x 64 float4
      int r = i >> 6, q = i & 63;
      *(float4*)(sA + r * 260 + q * 4) =
          *(const float4*)(X + (size_t)(mBase + r) * 512 + kc + q * 4);
      *(float4*)(sB + r * 260 + q * 4) =
          *(const float4*)(W + (size_t)(nBase + r) * 512 + kc + q * 4);
    }
    __syncthreads();

    const float* aP0 = sA + (mi0 * 16 + l16) * 260 + hi * 2;
    const float* bP0 = sB + (ni0 * 16 + l16) * 260 + hi * 2;
    const float* aP1 = sA + (mi1 * 16 + l16) * 260 + hi * 2;
    const float* bP1 = sB + (ni1 * 16 + l16) * 260 + hi * 2;
    #pragma unroll 4
    for (int k = 0; k < 256; k += 4) {
      v2f a0 = *(const v2f*)(aP0 + k);
      v2f b0 = *(const v2f*)(bP0 + k);
      acc0 = __builtin_amdgcn_wmma_f32_16x16x4_f32(false, a0, false, b0,
                                                   (short)0, acc0, false, false);
      v2f a1 = *(const v2f*)(aP1 + k);
      v2f b1v = *(const v2f*)(bP1 + k);
      acc1 = __builtin_amdgcn_wmma_f32_16x16x4_f32(false, a1, false, b1v,
                                                   (short)0, acc1, false, false);
    }
  }

  // C/D layout: VGPR r -> (M = r + hi*8, N = l16)
  {
    int N0 = nBase + ni0 * 16 + l16;
    float bias0 = b1[N0] + b2[N0];
    float* o0 = gx + (size_t)(mBase + mi0 * 16 + hi * 8) * G4H + N0;
    #pragma unroll
    for (int r = 0; r < 8; ++r) o0[(size_t)r * G4H] = acc0[r] + bias0;

    int N1 = nBase + ni1 * 16 + l16;
    float bias1 = b1[N1] + b2[N1];
    float* o1 = gx + (size_t)(mBase + mi1 * 16 + hi * 8) * G4H + N1;
    #pragma unroll
    for (int r = 0; r < 8; ++r) o1[(size_t)r * G4H] = acc1[r] + bias1;
  }
}

// ---------------------------------------------------------------------------
__global__ void bar_reset(unsigned* bar) {
  if (threadIdx.x == 0) { bar[0] = 0u; bar[1] = 0u; }
}

// ---------------------------------------------------------------------------
// Recurrent scan: 32 WGs, WG owns 16 h-columns (all 4 gates).  W_hh slice is
// LDS-resident for the whole kernel.  16 waves = 4 gates x 4 K-quarters;
// each wave runs 2 independent 16-deep v_wmma_f32_16x16x4_f32 chains
// (serial depth 16 instead of 64).  gx_t is loaded at the top of the step so
// its HBM latency overlaps the GEMM.  Device-wide atomic barrier per step;
// h ping-pongs through L2.
// ---------------------------------------------------------------------------
__global__ __launch_bounds__(512) void lstm_scan(
    const float* __restrict__ gx,     // (B,T,4H), both biases folded in
    const float* __restrict__ h0,     // (B,H) layer slice
    const float* __restrict__ c0,     // (B,H) layer slice
    const float* __restrict__ whh,    // (4H,H)
    float* __restrict__ out_seq,      // (B,T,H)
    float* __restrict__ h_bufs,       // 2*(B*H) ping-pong
    float* __restrict__ h_final,      // (B,H) slice of d_out
    unsigned* __restrict__ bar)       // [0]=count [1]=generation
{
  __shared__ float sW[64 * 516];      // 132 KB: 64 gate rows x 512 K (pad 4)
  __shared__ float sH[16 * 516];      //  33 KB: h_t staged
  __shared__ float sAcc[16 * 256];    //  16 KB: per-wave partial C tiles
  __shared__ float sC[256];           //   1 KB: cell state (m,j)

  const int tid  = threadIdx.x;
  const int lane = tid & 31;
  const int wave = tid >> 5;          // 0..15
  const int gate = wave >> 2;         // 0..3 -> i,f,g,o
  const int kq   = wave & 3;          // K quarter (128 wide)
  const int l16  = lane & 15;
  const int hi   = lane >> 4;
  const int colBase = blockIdx.x * 16;

  // Stage W_hh tile: local row (gate*16+n) <- global row gate*512+colBase+n
  for (int i = tid; i < 64 * 128; i += 512) {
    int r = i >> 7, q = i & 127;
    int g = r >> 4, n = r & 15;
    *(float4*)(sW + r * 516 + q * 4) =
        *(const float4*)(whh + (size_t)(g * LSTM_H + colBase + n) * LSTM_H + q * 4);
  }
  const bool doew = (tid < 256);                    // elementwise ownership
  const int  em = tid >> 4, ej = tid & 15;          // (batch m, col j)
  if (doew) sC[tid] = c0[em * LSTM_H + colBase + ej];

  // Fragment base pointers (wmma f32 16x16x4 A/B layouts, half-wave K pairs)
  const float* aP = sH + l16 * 516 + kq * 128 + hi * 2;
  const float* bP = sW + (gate * 16 + l16) * 516 + kq * 128 + hi * 2;
  const float* gpe = gx + (size_t)em * LSTM_T * G4H + colBase + ej;  // + t*G4H

  for (int t = 0; t < LSTM_T; ++t) {
    // Kick off gx_t loads early -> overlap HBM latency with the GEMM below.
    float xgi = 0.f, xgf = 0.f, xgg = 0.f, xgo = 0.f;
    if (doew) {
      const float* gp = gpe + (size_t)t * G4H;
      xgi = gp[0]; xgf = gp[512]; xgg = gp[1024]; xgo = gp[1536];
      __builtin_prefetch(gp + G4H, 0, 0);          // next timestep's slice
    }

    // stage h_t (all 512 cols)
    const float* hsrc = (t == 0) ? h0
                                 : (h_bufs + (size_t)(t & 1) * (LSTM_B * LSTM_H));
    for (int i = tid; i < 16 * 128; i += 512) {
      int m = i >> 7, q = i & 127;
      *(float4*)(sH + m * 516 + q * 4) =
          *(const float4*)(hsrc + (size_t)m * LSTM_H + q * 4);
    }
    __syncthreads();

    // Two independent accumulator chains over this wave's K quarter.
    v8f accA = {}; v8f accB = {};
    #pragma unroll 4
    for (int k = 0; k < 64; k += 4) {
      v2f a0 = *(const v2f*)(aP + k);
      v2f b0 = *(const v2f*)(bP + k);
      accA = __builtin_amdgcn_wmma_f32_16x16x4_f32(false, a0, false, b0,
                                                   (short)0, accA, false, false);
      v2f a1 = *(const v2f*)(aP + 64 + k);
      v2f b1 = *(const v2f*)(bP + 64 + k);
      accB = __builtin_amdgcn_wmma_f32_16x16x4_f32(false, a1, false, b1,
                                                   (short)0, accB, false, false);
    }
    {
      float* dst = sAcc + wave * 256 + hi * 8 * 16 + l16;
      #pragma unroll
      for (int r = 0; r < 8; ++r) dst[r * 16] = accA[r] + accB[r];
    }
    __syncthreads();

    // Cell update: threads 0..255 -> (batch m, column j)
    if (doew) {
      const float* A = sAcc + em * 16 + ej;
      float gi = A[0]    + A[256]  + A[512]  + A[768];
      float gf = A[1024] + A[1280] + A[1536] + A[1792];
      float gg = A[2048] + A[2304] + A[2560] + A[2816];
      float go = A[3072] + A[3328] + A[3584] + A[3840];
      gi = sigmoidf_(gi + xgi);
      gf = sigmoidf_(gf + xgf);
      gg = tanhf   (gg + xgg);
      go = sigmoidf_(go + xgo);
      float c = gf * sC[tid] + gi * gg;
      float h = go * tanhf(c);
      sC[tid] = c;
      h_bufs[(size_t)((t + 1) & 1) * (LSTM_B * LSTM_H) + em * LSTM_H + colBase + ej] = h;
      out_seq[((size_t)em * LSTM_T + t) * LSTM_H + colBase + ej] = h;
      if (t == LSTM_T - 1) h_final[em * LSTM_H + colBase + ej] = h;
    }

    // Device-wide barrier: h_{t+1} must be visible to all 32 WGs.
    __threadfence();
    __syncthreads();
    if (tid == 0) {
      unsigned g = __hip_atomic_load(bar + 1, __ATOMIC_ACQUIRE, __HIP_MEMORY_SCOPE_AGENT);
      unsigned prev = __hip_atomic_fetch_add(bar, 1u, __ATOMIC_ACQ_REL, __HIP_MEMORY_SCOPE_AGENT);
      if (prev == NWG - 1) {
        __hip_atomic_store(bar, 0u, __ATOMIC_RELAXED, __HIP_MEMORY_SCOPE_AGENT);
        __hip_atomic_fetch_add(bar + 1, 1u, __ATOMIC_RELEASE, __HIP_MEMORY_SCOPE_AGENT);
      } else {
        while (__hip_atomic_load(bar + 1, __ATOMIC_ACQUIRE, __HIP_MEMORY_SCOPE_AGENT) == g) {
          __builtin_amdgcn_s_sleep(1);
        }
      }
    }
    __syncthreads();
  }
}

// ---------------------------------------------------------------------------
extern "C" void kernel_launch(void* const* d_in, const int* in_sizes, int n_in,
                              void* d_out, int out_size, void* d_ws, size_t ws_size,
                              hipStream_t stream) {
  (void)in_sizes; (void)n_in; (void)out_size; (void)ws_size;
  const float* x    = (const float*)d_in[0];
  const float* h0   = (const float*)d_in[1];   // (2,B,H)
  const float* c0   = (const float*)d_in[2];   // (2,B,H)
  const float* wih0 = (const float*)d_in[3];
  const float* whh0 = (const float*)d_in[4];
  const float* bih0 = (const float*)d_in[5];
  const float* bhh0 = (const float*)d_in[6];
  const float* wih1 = (const float*)d_in[7];
  const float* whh1 = (const float*)d_in[8];
  const float* bih1 = (const float*)d_in[9];
  const float* bhh1 = (const float*)d_in[10];
  float* out = (float*)d_out;                  // (2,B,H)

  // workspace layout (needs ~161 MB):
  float*    gx   = (float*)d_ws;                              // B*T*4H  (128 MB)
  float*    out0 = gx + (size_t)LSTM_B * LSTM_T * G4H;        // B*T*H   (32 MB)
  float*    hbuf = out0 + (size_t)LSTM_B * LSTM_T * LSTM_H;   // 2*B*H
  unsigned* bar  = (unsigned*)(hbuf + 2 * LSTM_B * LSTM_H);   // counter+gen

  dim3 pgrid(G4H / 64, (LSTM_B * LSTM_T) / 64);

  // ---- layer 0 ----
  proj_gemm<<<pgrid, 256, 0, stream>>>(x, wih0, bih0, bhh0, gx);
  bar_reset<<<1, 32, 0, stream>>>(bar);
  lstm_scan<<<NWG, 512, 0, stream>>>(gx, h0, c0, whh0, out0, hbuf, out, bar);
  // ---- layer 1 ----
  proj_gemm<<<pgrid, 256, 0, stream>>>(out0, wih1, bih1, bhh1, gx);
  bar_reset<<<1, 32, 0, stream>>>(bar);
  lstm_scan<<<NWG, 512, 0, stream>>>(gx, h0 + LSTM_B * LSTM_H, c0 + LSTM_B * LSTM_H,
                                     whh1, out0, hbuf, out + LSTM_B * LSTM_H, bar);
}